// GraphTransformer_28329604284667
// MI455X (gfx1250) — compile-verified
//
#include <hip/hip_runtime.h>
#include <hip/hip_bf16.h>
#include <math.h>

#define NH     4
#define DH     32
#define NNODES 50000
#define NEDGES 800000
#define CINC   256
#define COUTC  128

typedef float v2f __attribute__((ext_vector_type(2)));
typedef float v8f __attribute__((ext_vector_type(8)));

__device__ __forceinline__ float atomAddF(float* p, float v) {
  return __hip_atomic_fetch_add(p, v, __ATOMIC_RELAXED, __HIP_MEMORY_SCOPE_AGENT);
}

// Float atomic max via signed/unsigned int ordering trick (works for mixed signs,
// init value -inf = 0xFF800000).
__device__ __forceinline__ void atomMaxF(float* addr, float value) {
  if (value >= 0.0f)
    atomicMax((int*)addr, __float_as_int(value));
  else
    atomicMin((unsigned int*)addr, __float_as_uint(value));
}

// ---------------------------------------------------------------------------
// y0 = concat([x, x*t], axis=-1)   (N, 256)
// ---------------------------------------------------------------------------
__global__ void k_y0(const float* __restrict__ x, const float* __restrict__ t,
                     float* __restrict__ y0) {
  const long i = (long)blockIdx.x * blockDim.x + threadIdx.x;
  if (i >= (long)NNODES * CINC) return;
  const int n = (int)(i >> 8), c = (int)(i & 255);
  const float xv = x[(long)n * COUTC + (c & 127)];
  y0[i] = (c < 128) ? xv : xv * t[n];
}

// ---------------------------------------------------------------------------
// C(nrows x M) = A(nrows x K) @ W(K x M) + bias via V_WMMA_F32_16X16X4_F32.
// Block = 256 threads = 8 waves = ROWT x COLT 16x16 tiles. The A panel
// (ROWT*16 rows x K) is staged once per block into LDS with
// GLOBAL_LOAD_ASYNC_TO_LDS_B128 (ASYNCcnt) and shared by all waves.
// LDS row stride = K+4 floats: 16B-aligned rows, 4 banks apart, so the
// 32-lane ds_load_b64 A-fragment read covers all 64 banks conflict-free.
// A 16x4 frag: lanes 0-15 hold K=+0,+1 ; lanes 16-31 hold K=+2,+3 (ISA 7.12.2).
// ---------------------------------------------------------------------------
template <int K, int M, int ROWT, int COLT>   // ROWT*COLT == 8 waves
__global__ void __launch_bounds__(256)
k_gemm_wmma(const float* __restrict__ A, const float* __restrict__ W,
            const float* __restrict__ bias, float* __restrict__ C, int nrows) {
  constexpr int LDSK = K + 4;
  __shared__ float tileA[ROWT * 16 * LDSK];

  constexpr int mTiles = M / 16;
  constexpr int colBlocks = mTiles / COLT;
  const int rowBlock = blockIdx.x / colBlocks;
  const int colBlock = blockIdx.x - rowBlock * colBlocks;
  const int rowBase = rowBlock * ROWT * 16;

  // ---- stage A panel into LDS (async, 16B per lane per issue) ----
  constexpr int vecPerRow = K / 4;
  constexpr int totalVec = ROWT * 16 * vecPerRow;
  for (int i = threadIdx.x; i < totalVec; i += 256) {
    const int r = i / vecPerRow;
    const int c = i - r * vecPerRow;
    if (rowBase + r < nrows) {                       // per-lane EXEC mask is fine
      const float* gsrc = A + (long)(rowBase + r) * K + c * 4;
      // low 32 bits of the generic LDS address == wave-relative LDS offset
      const unsigned loff = (unsigned)(uintptr_t)&tileA[r * LDSK + c * 4];
      asm volatile("global_load_async_to_lds_b128 %0, %1, off"
                   :: "v"(loff), "v"(gsrc) : "memory");
    }
  }
  asm volatile("s_wait_asynccnt 0x0" ::: "memory");   // my copies landed
  __syncthreads();                                    // everyone's copies landed

  // ---- compute: one 16x16 tile per wave ----
  const int lane = threadIdx.x & 31;
  const int wave = threadIdx.x >> 5;
  const int rowIdx = wave / COLT;                     // 0..ROWT-1
  const int colIdx = wave - rowIdx * COLT;            // 0..COLT-1
  const int half = lane >> 4;                         // 0: lanes 0-15, 1: 16-31
  const int l = lane & 15;
  const int col = (colBlock * COLT + colIdx) * 16 + l;
  const float* As = &tileA[(rowIdx * 16 + l) * LDSK];

  v8f acc = {};
  for (int kk = 0; kk < K; kk += 4) {
    const int ka = kk + 2 * half;
    v2f a, b;
    a.x = As[ka];
    a.y = As[ka + 1];
    b.x = W[ka * M + col];
    b.y = W[(ka + 1) * M + col];
    acc = __builtin_amdgcn_wmma_f32_16x16x4_f32(false, a, false, b,
                                                (short)0, acc, false, false);
  }

  const int tileRow0 = rowBase + rowIdx * 16;
  if (tileRow0 < nrows) {                             // wave-uniform (N % 16 == 0)
    const float bv = bias[col];
    const long base = (long)(tileRow0 + 8 * half) * M + col;
#pragma unroll
    for (int r = 0; r < 8; ++r)
      C[base + (long)r * M] = acc[r] + bv;
  }
}

// ---------------------------------------------------------------------------
// Init segment-max to -inf, denom and agg to 0.
// ---------------------------------------------------------------------------
__global__ void k_init(float* __restrict__ m, float* __restrict__ denom,
                       float* __restrict__ agg) {
  const long i = (long)blockIdx.x * blockDim.x + threadIdx.x;
  if (i < (long)NNODES * NH) { m[i] = -INFINITY; denom[i] = 0.0f; }
  if (i < (long)NNODES * 128) agg[i] = 0.0f;
}

// ---------------------------------------------------------------------------
// Pass A: alpha[e,h] = <q[dst], k[src]+e> / sqrt(D);  segment max into m.
// One wave per edge, lane = dim within head.
// ---------------------------------------------------------------------------
__global__ void k_edge_alpha(const float* __restrict__ q, const float* __restrict__ kmat,
                             const int* __restrict__ src, const int* __restrict__ dst,
                             const float* __restrict__ ew,
                             const float* __restrict__ We, const float* __restrict__ be,
                             float* __restrict__ alpha, float* __restrict__ m) {
  const int lane = threadIdx.x & 31;
  const int e = blockIdx.x * (blockDim.x >> 5) + (threadIdx.x >> 5);
  if (e >= NEDGES) return;
  const int s = src[e], d = dst[e];
  const float w = ew[e];
  const float rsq = 0.17677669529663687f;   // 1/sqrt(32)
#pragma unroll
  for (int h = 0; h < NH; ++h) {
    const int o = h * DH + lane;
    const float ev = w * We[o] + be[o];
    const float qv = q[(long)d * 128 + o];
    const float kv = kmat[(long)s * 128 + o] + ev;
    float sum = qv * kv;
#pragma unroll
    for (int off = 16; off > 0; off >>= 1) sum += __shfl_xor(sum, off, 32);
    if (lane == 0) {
      const float a = sum * rsq;
      alpha[(long)e * NH + h] = a;
      atomMaxF(&m[d * NH + h], a);
    }
  }
}

// m = where(isfinite(m), m, 0)
__global__ void k_fix_m(float* __restrict__ m) {
  const int i = blockIdx.x * blockDim.x + threadIdx.x;
  if (i >= NNODES * NH) return;
  const float v = m[i];
  if (!(v > -INFINITY) || !(v < INFINITY)) m[i] = 0.0f;
}

// ---------------------------------------------------------------------------
// Pass B: p = exp(alpha - m[dst]); denom = segment_sum(p). (p overwrites alpha)
// ---------------------------------------------------------------------------
__global__ void k_edge_exp(const float* __restrict__ alpha, const int* __restrict__ dst,
                           const float* __restrict__ m, float* __restrict__ p,
                           float* __restrict__ denom) {
  const long i = (long)blockIdx.x * blockDim.x + threadIdx.x;
  if (i >= (long)NEDGES * NH) return;
  const int e = (int)(i >> 2);
  const int h = (int)(i & 3);
  const int d = dst[e];
  const float pe = expf(alpha[i] - m[d * NH + h]);
  p[i] = pe;
  atomAddF(&denom[d * NH + h], pe);
}

// ---------------------------------------------------------------------------
// Pass C: msg = (v[src]+e) * p/(denom[dst]+1e-16); agg = segment_sum(msg).
// One wave per edge, lane = dim within head.
// ---------------------------------------------------------------------------
__global__ void k_edge_msg(const float* __restrict__ vmat, const int* __restrict__ src,
                           const int* __restrict__ dst, const float* __restrict__ ew,
                           const float* __restrict__ We, const float* __restrict__ be,
                           const float* __restrict__ p, const float* __restrict__ denom,
                           float* __restrict__ agg) {
  const int lane = threadIdx.x & 31;
  const int e = blockIdx.x * (blockDim.x >> 5) + (threadIdx.x >> 5);
  if (e >= NEDGES) return;
  const int s = src[e], d = dst[e];
  const float w = ew[e];
#pragma unroll
  for (int h = 0; h < NH; ++h) {
    const int o = h * DH + lane;
    const float attn = p[(long)e * NH + h] / (denom[d * NH + h] + 1e-16f);
    const float ev = w * We[o] + be[o];
    const float msg = (vmat[(long)s * 128 + o] + ev) * attn;
    atomAddF(&agg[(long)d * 128 + o], msg);
  }
}

// y1 = tanh(mean_h(agg) + skip)    (N, 32)
__global__ void k_mean_tanh(const float* __restrict__ agg, const float* __restrict__ skip,
                            float* __restrict__ y1) {
  const int i = blockIdx.x * blockDim.x + threadIdx.x;
  if (i >= NNODES * DH) return;
  const int n = i >> 5, dd = i & 31;
  const long b = (long)n * 128 + dd;
  const float s = 0.25f * (agg[b] + agg[b + 32] + agg[b + 64] + agg[b + 96]);
  y1[i] = tanhf(s + skip[i]);
}

// out = x * tanh(y2[:, :128]) + tanh(y2[:, 128:])
__global__ void k_final(const float* __restrict__ x, const float* __restrict__ y2,
                        float* __restrict__ out) {
  const long i = (long)blockIdx.x * blockDim.x + threadIdx.x;
  if (i >= (long)NNODES * COUTC) return;
  const int n = (int)(i >> 7), c = (int)(i & 127);
  out[i] = x[i] * tanhf(y2[(long)n * 256 + c]) + tanhf(y2[(long)n * 256 + 128 + c]);
}

// ---------------------------------------------------------------------------
extern "C" void kernel_launch(void* const* d_in, const int* in_sizes, int n_in,
                              void* d_out, int out_size, void* d_ws, size_t ws_size,
                              hipStream_t stream) {
  (void)in_sizes; (void)n_in; (void)out_size; (void)ws_size;
  const float* x     = (const float*)d_in[0];
  const float* t     = (const float*)d_in[1];
  const int*   ei    = (const int*)d_in[2];
  const float* ew    = (const float*)d_in[3];
  const float* Wq    = (const float*)d_in[4];
  const float* bq    = (const float*)d_in[5];
  const float* Wk    = (const float*)d_in[6];
  const float* bk    = (const float*)d_in[7];
  const float* Wv    = (const float*)d_in[8];
  const float* bv    = (const float*)d_in[9];
  const float* We    = (const float*)d_in[10];
  const float* be    = (const float*)d_in[11];
  const float* Wskip = (const float*)d_in[12];
  const float* bskip = (const float*)d_in[13];
  const float* Wmlp  = (const float*)d_in[14];
  const float* bmlp  = (const float*)d_in[15];
  float* out = (float*)d_out;
  const int* src = ei;
  const int* dst = ei + NEDGES;

  // Workspace layout (floats). y2 aliases y0, y1 aliases q (both dead by then).
  float* ws    = (float*)d_ws;
  float* y0    = ws;                          // N*256
  float* q     = y0    + (long)NNODES * 256;  // N*128
  float* kmat  = q     + (long)NNODES * 128;  // N*128
  float* vmat  = kmat  + (long)NNODES * 128;  // N*128
  float* skip  = vmat  + (long)NNODES * 128;  // N*32
  float* m     = skip  + (long)NNODES * 32;   // N*4
  float* denom = m     + (long)NNODES * 4;    // N*4
  float* p     = denom + (long)NNODES * 4;    // E*4 (alpha, then softmax numerator)
  float* agg   = p     + (long)NEDGES * 4;    // N*128
  float* y1    = q;                           // alias: N*32
  float* y2    = y0;                          // alias: N*256

  const dim3 B(256);
  const int WPB = 8;

  k_y0<<<(NNODES * CINC + 255) / 256, B, 0, stream>>>(x, t, y0);

  // Q/K/V: K=256, M=128 -> 8 col tiles = one block row-panel, ROWT=1/COLT=8.
  const int nTiles = NNODES / 16;             // 3125
  k_gemm_wmma<256, 128, 1, 8><<<nTiles, B, 0, stream>>>(y0, Wq, bq, q,    NNODES);
  k_gemm_wmma<256, 128, 1, 8><<<nTiles, B, 0, stream>>>(y0, Wk, bk, kmat, NNODES);
  k_gemm_wmma<256, 128, 1, 8><<<nTiles, B, 0, stream>>>(y0, Wv, bv, vmat, NNODES);
  // Skip: K=256, M=32 -> COLT=2, ROWT=4 (partial last block handled by guards).
  k_gemm_wmma<256, 32, 4, 2><<<(nTiles + 3) / 4, B, 0, stream>>>(y0, Wskip, bskip, skip, NNODES);

  k_init<<<(NNODES * 128 + 255) / 256, B, 0, stream>>>(m, denom, agg);

  k_edge_alpha<<<(NEDGES + WPB - 1) / WPB, B, 0, stream>>>(q, kmat, src, dst, ew, We, be, p, m);
  k_fix_m<<<(NNODES * NH + 255) / 256, B, 0, stream>>>(m);
  k_edge_exp<<<(NEDGES * NH + 255) / 256, B, 0, stream>>>(p, dst, m, p, denom);
  k_edge_msg<<<(NEDGES + WPB - 1) / WPB, B, 0, stream>>>(vmat, src, dst, ew, We, be, p, denom, agg);

  k_mean_tanh<<<(NNODES * DH + 255) / 256, B, 0, stream>>>(agg, skip, y1);

  // MLP: K=32, M=256 -> 16 col tiles -> colBlocks=2, ROWT=1/COLT=8.
  k_gemm_wmma<32, 256, 1, 8><<<nTiles * 2, B, 0, stream>>>(y1, Wmlp, bmlp, y2, NNODES);

  k_final<<<(NNODES * COUTC + 255) / 256, B, 0, stream>>>(x, y2, out);
}